// Drug_Molecular_9440338117056
// MI455X (gfx1250) — compile-verified
//
#include <hip/hip_runtime.h>

// ---------------- problem constants (fixed by the reference) ----------------
#define NN      30000      // nodes
#define E0      480000     // raw edges
#define ET      510000     // edges + self loops
#define F0      256        // input feature dim
#define H1      4          // layer-1 heads
#define D1      128        // layer-1 per-head dim
#define HD1     512        // H1*D1
#define C2      256        // layer-2 dim (1 head x 256)
#define NG      64         // graphs
#define EPSBN   1e-5f
#define FMAXF   3.402823466e+38f

typedef __attribute__((ext_vector_type(2))) float v2f;
typedef __attribute__((ext_vector_type(8))) float v8f;

// Monotone float<->int encoding so signed-int atomicMax == float max.
__device__ __forceinline__ int fenc(float f) {
  int i = __float_as_int(f);
  return i >= 0 ? i : (i ^ 0x7FFFFFFF);
}
__device__ __forceinline__ float fdec(int i) {
  return __int_as_float(i >= 0 ? i : (i ^ 0x7FFFFFFF));
}

// ---------------- fills / copies ----------------
__global__ void fill_i32(int* __restrict__ p, int v, int n) {
  int t = blockIdx.x * blockDim.x + threadIdx.x;
  if (t < n) p[t] = v;
}
__global__ void copy_i32(const int* __restrict__ a, int* __restrict__ b, int n) {
  int t = blockIdx.x * blockDim.x + threadIdx.x;
  if (t < n) b[t] = a[t];
}

// ================= CSR build (dst-sorted adjacency, built once) =============
__global__ void count_deg(const int* __restrict__ dstE, int* __restrict__ cnt) {
  int t = blockIdx.x * blockDim.x + threadIdx.x;
  if (t >= ET) return;
  int d = (t < E0) ? dstE[t] : (t - E0);
  atomicAdd(&cnt[d], 1);
}

// single-block Hillis-Steele scan over NN counts -> exclusive rowptr[NN+1]
__global__ void scan_rowptr(const int* __restrict__ cnt, int* __restrict__ rowptr, int n) {
  __shared__ int buf[1024];
  __shared__ int carry;
  if (threadIdx.x == 0) carry = 0;
  __syncthreads();
  for (int base = 0; base < n; base += 1024) {
    int i = base + threadIdx.x;
    int v = (i < n) ? cnt[i] : 0;
    buf[threadIdx.x] = v;
    __syncthreads();
    for (int off = 1; off < 1024; off <<= 1) {
      int add = (threadIdx.x >= (unsigned)off) ? buf[threadIdx.x - off] : 0;
      __syncthreads();
      buf[threadIdx.x] += add;
      __syncthreads();
    }
    if (i < n) rowptr[i + 1] = carry + buf[threadIdx.x];
    __syncthreads();
    if (threadIdx.x == 0) carry += buf[1023];
    __syncthreads();
  }
  if (threadIdx.x == 0) rowptr[0] = 0;
}

__global__ void scatter_csr(const int* __restrict__ srcE, const int* __restrict__ dstE,
                            int* __restrict__ cursor, int* __restrict__ col) {
  int t = blockIdx.x * blockDim.x + threadIdx.x;
  if (t >= ET) return;
  int s = (t < E0) ? srcE[t] : (t - E0);
  int d = (t < E0) ? dstE[t] : (t - E0);
  int pos = atomicAdd(&cursor[d], 1);
  col[pos] = s;
}

// ---------------- fp32 WMMA GEMM: C[M,N] = A[M,K] @ B[K,N] ----------------
// blockIdx.x = M-tile; wave w computes the 16x64 strip at columns w*64.
// The 16xK A panel is contiguous in global memory -> coalesced float4 stage
// into LDS (row stride K+4 to spread banks), shared by all waves in block.
// No divergent branches -> EXEC all-ones at every WMMA (ISA requirement).
__global__ void gemm_wmma_f32(const float* __restrict__ A,
                              const float* __restrict__ B,
                              float* __restrict__ C,
                              int K, int N) {
  extern __shared__ float As[];                 // 16 * (K+4) floats
  const int mTile = blockIdx.x;
  const int lane  = threadIdx.x & 31;
  const int wave  = threadIdx.x >> 5;           // = nGroup (N/64 waves/block)
  const int Ks = K + 4;                         // padded LDS row stride

  // cooperative stage: 16*K contiguous floats, float4-coalesced
  const float* Ablk = A + (size_t)mTile * 16 * K;
  const int total4 = (16 * K) >> 2;
  for (int i4 = threadIdx.x; i4 < total4; i4 += blockDim.x) {
    int fl = i4 << 2;
    int row = fl / K, k = fl - row * K;
    *(float4*)(As + row * Ks + k) = ((const float4*)Ablk)[i4];
  }
  __syncthreads();

  const int half = lane >> 4;                   // 0: K pair {0,1}, 1: {2,3}
  const int lr   = lane & 15;
  const int colBase = wave * 64 + lr;
  const float* Asrow = As + lr * Ks;

  v8f acc0 = {}, acc1 = {}, acc2 = {}, acc3 = {};
  for (int k0 = 0; k0 < K; k0 += 4) {
    const int ka = k0 + 2 * half;
    v2f a; a.x = Asrow[ka]; a.y = Asrow[ka + 1];        // ds_load_b64
    const float* B0 = B + (size_t)ka * N;               // L2-resident weights
    const float* B1 = B0 + N;
    v2f b0, b1, b2, b3;
    b0.x = B0[colBase];      b0.y = B1[colBase];
    b1.x = B0[colBase + 16]; b1.y = B1[colBase + 16];
    b2.x = B0[colBase + 32]; b2.y = B1[colBase + 32];
    b3.x = B0[colBase + 48]; b3.y = B1[colBase + 48];
    acc0 = __builtin_amdgcn_wmma_f32_16x16x4_f32(false, a, false, b0, (short)0, acc0, false, false);
    acc1 = __builtin_amdgcn_wmma_f32_16x16x4_f32(false, a, false, b1, (short)0, acc1, false, false);
    acc2 = __builtin_amdgcn_wmma_f32_16x16x4_f32(false, a, false, b2, (short)0, acc2, false, false);
    acc3 = __builtin_amdgcn_wmma_f32_16x16x4_f32(false, a, false, b3, (short)0, acc3, false, false);
  }
  // C/D layout: VGPR r -> M = r + 8*half, N = lr (ISA 16x16 f32 table)
  #pragma unroll
  for (int r = 0; r < 8; ++r) {
    float* Crow = C + (size_t)(mTile * 16 + r + 8 * half) * N + wave * 64 + lr;
    Crow[0]  = acc0[r];
    Crow[16] = acc1[r];
    Crow[32] = acc2[r];
    Crow[48] = acc3[r];
  }
}

// ---------------- per-(node,head) attention dot products ----------------
__global__ void attn_dots(const float* __restrict__ h,
                          const float* __restrict__ att_s,
                          const float* __restrict__ att_d,
                          float* __restrict__ as_, float* __restrict__ ad_,
                          int N, int H, int D) {
  const int wave = (blockIdx.x * blockDim.x + threadIdx.x) >> 5;
  const int lane = threadIdx.x & 31;
  if (wave >= N * H) return;
  const int n = wave / H, hh = wave - n * H;
  const float* hp = h + ((size_t)n * H + hh) * D;
  const float* sp = att_s + hh * D;
  const float* dp = att_d + hh * D;
  float s = 0.f, d = 0.f;
  for (int j = lane; j < D; j += 32) { float v = hp[j]; s += v * sp[j]; d += v * dp[j]; }
  for (int off = 16; off; off >>= 1) { s += __shfl_down(s, off, 32); d += __shfl_down(d, off, 32); }
  if (lane == 0) { as_[wave] = s; ad_[wave] = d; }
}

// ======= fused dst-centric segment-softmax + weighted aggregation ==========
// One wave per (node, head). Pass 1: lane-split max + butterfly reduce.
// Pass 2: stream incident edges; w = exp(e-m); acc += w * h_src (lanes split
// D, coalesced 128B row gathers); out = acc / sum(w). Zero atomics.
__global__ void gat_csr(const int* __restrict__ rowptr, const int* __restrict__ col,
                        const float* __restrict__ as_, const float* __restrict__ ad_,
                        const float* __restrict__ h, float* __restrict__ outp,
                        int H, int D) {
  const int wave = (blockIdx.x * blockDim.x + threadIdx.x) >> 5;
  const int lane = threadIdx.x & 31;
  if (wave >= NN * H) return;
  const int n = wave / H, hh = wave - n * H;
  const int beg = rowptr[n], end = rowptr[n + 1];
  const float adn = ad_[n * H + hh];

  float m = -FMAXF;
  for (int i = beg + lane; i < end; i += 32) {
    float v = as_[col[i] * H + hh] + adn;
    m = fmaxf(m, v > 0.f ? v : 0.2f * v);          // leaky_relu slope 0.2
  }
  for (int off = 16; off; off >>= 1) m = fmaxf(m, __shfl_xor(m, off, 32));

  float acc[8];
  const int nj = D >> 5;                            // 4 (L1) or 8 (L2)
  #pragma unroll
  for (int j = 0; j < 8; ++j) acc[j] = 0.f;
  float den = 0.f;
  for (int i = beg; i < end; ++i) {
    int s = col[i];
    float v = as_[s * H + hh] + adn;
    v = v > 0.f ? v : 0.2f * v;
    float w = __expf(v - m);
    den += w;
    const float* hp = h + ((size_t)s * H + hh) * D + lane;
    for (int j = 0; j < nj; ++j) acc[j] += w * hp[j * 32];
  }
  float inv = 1.f / den;                            // >=1 self-loop -> den>0
  float* op = outp + ((size_t)n * H + hh) * D + lane;
  for (int j = 0; j < nj; ++j) op[j * 32] = acc[j] * inv;
}

// ---------------- bias + optional leaky relu ----------------
__global__ void bias_act(float* __restrict__ x, const float* __restrict__ b,
                         int n, int C, float slope) {
  int t = blockIdx.x * blockDim.x + threadIdx.x;
  if (t >= n) return;
  int c = t % C;
  float v = x[t] + b[c];
  if (slope != 1.f) v = v > 0.f ? v : slope * v;
  x[t] = v;
}

// ---------------- batch norm: per-column stats then apply ----------------
__global__ void bn_stats(const float* __restrict__ x, float* __restrict__ sums,
                         float* __restrict__ sqs, int N, int C) {
  __shared__ float ss[256], qq[256];
  int c = blockIdx.x;
  float s = 0.f, q = 0.f;
  for (int r = threadIdx.x; r < N; r += blockDim.x) {
    float v = x[(size_t)r * C + c]; s += v; q += v * v;
  }
  ss[threadIdx.x] = s; qq[threadIdx.x] = q;
  __syncthreads();
  for (int st = blockDim.x >> 1; st; st >>= 1) {
    if (threadIdx.x < (unsigned)st) { ss[threadIdx.x] += ss[threadIdx.x + st]; qq[threadIdx.x] += qq[threadIdx.x + st]; }
    __syncthreads();
  }
  if (threadIdx.x == 0) { sums[c] = ss[0]; sqs[c] = qq[0]; }
}

__global__ void bn_apply(float* __restrict__ x, const float* __restrict__ sums,
                         const float* __restrict__ sqs, const float* __restrict__ gamma,
                         const float* __restrict__ beta, int N, int C) {
  int t = blockIdx.x * blockDim.x + threadIdx.x;
  if (t >= N * C) return;
  int c = t % C;
  float mu  = sums[c] / (float)N;
  float var = sqs[c] / (float)N - mu * mu;     // biased variance (PyTorch norm)
  x[t] = gamma[c] * (x[t] - mu) * rsqrtf(var + EPSBN) + beta[c];
}

// ---------------- per-graph max pool ----------------
__global__ void pool_max(const float* __restrict__ x, const int* __restrict__ batch,
                         int* __restrict__ pool, int N, int C) {
  int t = blockIdx.x * blockDim.x + threadIdx.x;
  if (t >= N * C) return;
  int n = t / C, c = t - n * C;
  atomicMax(&pool[batch[n] * C + c], fenc(x[t]));
}
__global__ void pool_decode(const int* __restrict__ pool, float* __restrict__ out, int n) {
  int t = blockIdx.x * blockDim.x + threadIdx.x;
  if (t < n) out[t] = fdec(pool[t]);
}

static inline int cdiv(long long a, long long b) { return (int)((a + b - 1) / b); }

extern "C" void kernel_launch(void* const* d_in, const int* in_sizes, int n_in,
                              void* d_out, int out_size, void* d_ws, size_t ws_size,
                              hipStream_t stream) {
  (void)in_sizes; (void)n_in; (void)out_size; (void)ws_size;
  const float* x        = (const float*)d_in[0];
  const int*   srcE     = (const int*)d_in[1];          // edge_index[0]
  const int*   dstE     = srcE + E0;                    // edge_index[1]
  const int*   batch    = (const int*)d_in[2];
  const float* W1       = (const float*)d_in[3];
  const float* att_src1 = (const float*)d_in[4];
  const float* att_dst1 = (const float*)d_in[5];
  const float* b1       = (const float*)d_in[6];
  const float* gamma1   = (const float*)d_in[7];
  const float* beta1    = (const float*)d_in[8];
  const float* W2       = (const float*)d_in[9];
  const float* att_src2 = (const float*)d_in[10];
  const float* att_dst2 = (const float*)d_in[11];
  const float* b2       = (const float*)d_in[12];
  const float* gamma2   = (const float*)d_in[13];
  const float* beta2    = (const float*)d_in[14];
  float* out = (float*)d_out;

  // ---- workspace layout (floats) ----
  float* ws = (float*)d_ws;
  size_t o = 0;
  float* h1   = ws + o; o += (size_t)NN * HD1;   // GEMM1 output   (61.4 MB)
  float* agg1 = ws + o; o += (size_t)NN * HD1;   // layer-1 agg    (61.4 MB)
  float* as1  = ws + o; o += (size_t)NN * H1;
  float* ad1  = ws + o; o += (size_t)NN * H1;
  int*   rowptr = (int*)(ws + o); o += NN + 1;
  int*   cursor = (int*)(ws + o); o += NN;       // counts, then scatter cursor
  int*   colx   = (int*)(ws + o); o += ET;       // CSR src ids    (2.0 MB)
  float* csum = ws + o; o += 512;
  float* csq  = ws + o; o += 512;
  int*   pool = (int*)(ws + o); o += (size_t)NG * C2;
  // layer-2 aliases (h1 dead after layer-1 aggregation; stream-ordered)
  float* h2   = h1;
  float* agg2 = h1 + (size_t)NN * C2;
  float* as2 = as1; float* ad2 = ad1;

  const int T = 256;
  // host-side fenc(-FLT_MAX): bits 0xFF7FFFFF ^ 0x7FFFFFFF = 0x80800000
  const int ENC_NEG_INF = (int)0x80800000;

  // ================= CSR build (once, reused by both layers) ===============
  fill_i32<<<cdiv(NN, T), T, 0, stream>>>(cursor, 0, NN);
  count_deg<<<cdiv(ET, T), T, 0, stream>>>(dstE, cursor);
  scan_rowptr<<<1, 1024, 0, stream>>>(cursor, rowptr, NN);
  copy_i32<<<cdiv(NN, T), T, 0, stream>>>(rowptr, cursor, NN);
  scatter_csr<<<cdiv(ET, T), T, 0, stream>>>(srcE, dstE, cursor, colx);

  // ================= Layer 1 =================
  // GEMM1: h1[30000,512] = x @ W1 ; 1875 blocks x 8 waves, 16KB LDS A-panel
  gemm_wmma_f32<<<1875, 256, 16 * (F0 + 4) * 4, stream>>>(x, W1, h1, F0, HD1);
  attn_dots<<<cdiv((long long)NN * H1 * 32, T), T, 0, stream>>>(h1, att_src1, att_dst1, as1, ad1, NN, H1, D1);
  gat_csr<<<cdiv((long long)NN * H1 * 32, T), T, 0, stream>>>(rowptr, colx, as1, ad1, h1, agg1, H1, D1);
  bias_act<<<cdiv((long long)NN * HD1, T), T, 0, stream>>>(agg1, b1, NN * HD1, HD1, 0.01f);
  bn_stats<<<HD1, T, 0, stream>>>(agg1, csum, csq, NN, HD1);
  bn_apply<<<cdiv((long long)NN * HD1, T), T, 0, stream>>>(agg1, csum, csq, gamma1, beta1, NN, HD1);

  // ================= Layer 2 =================
  // GEMM2: h2[30000,256] = agg1 @ W2 ; 1875 blocks x 4 waves, 33KB LDS panel
  gemm_wmma_f32<<<1875, 128, 16 * (HD1 + 4) * 4, stream>>>(agg1, W2, h2, HD1, C2);
  attn_dots<<<cdiv((long long)NN * 32, T), T, 0, stream>>>(h2, att_src2, att_dst2, as2, ad2, NN, 1, C2);
  gat_csr<<<cdiv((long long)NN * 32, T), T, 0, stream>>>(rowptr, colx, as2, ad2, h2, agg2, 1, C2);
  bias_act<<<cdiv((long long)NN * C2, T), T, 0, stream>>>(agg2, b2, NN * C2, C2, 1.0f);
  bn_stats<<<C2, T, 0, stream>>>(agg2, csum, csq, NN, C2);
  bn_apply<<<cdiv((long long)NN * C2, T), T, 0, stream>>>(agg2, csum, csq, gamma2, beta2, NN, C2);

  // ================= Global max pool per graph =================
  fill_i32<<<cdiv(NG * C2, T), T, 0, stream>>>(pool, ENC_NEG_INF, NG * C2);
  pool_max<<<cdiv((long long)NN * C2, T), T, 0, stream>>>(agg2, batch, pool, NN, C2);
  pool_decode<<<cdiv(NG * C2, T), T, 0, stream>>>(pool, out, NG * C2);
}